// GSN_35433480192471
// MI455X (gfx1250) — compile-verified
//
#include <hip/hip_runtime.h>

#define N_NODES     100000
#define N_EDGES_IN  1600000
#define N_EDGES_TOT 1700000   // + self loops
#define N_DIM       64
#define N_FEAT      32
#define N_LEV       4
#define N_STEPS     4
#define NF          (N_NODES * N_FEAT)   // 3,200,000
#define NKP         (N_DIM / 2)          // 32 K-pairs per level

typedef __attribute__((ext_vector_type(2))) float v2f;
typedef __attribute__((ext_vector_type(8))) float v8f;

__device__ __forceinline__ void edge_pair(const int* __restrict__ ei, int e, int& d, int& s) {
    if (e < N_EDGES_IN) { d = ei[2 * e]; s = ei[2 * e + 1]; }
    else { d = e - N_EDGES_IN; s = d; }
}

// ---- init: Y[0]=1 in both buffers, Ya levels 1..4 = 0, deg = 0 --------------
__global__ void k_init(float* __restrict__ Ya, float* __restrict__ Yb,
                       float* __restrict__ deg) {
    int i = blockIdx.x * blockDim.x + threadIdx.x;
    if (i < NF)          { Ya[i] = 1.0f; Yb[i] = 1.0f; }
    else if (i < 5 * NF) { Ya[i] = 0.0f; }
    if (i < N_NODES) deg[i] = 0.0f;
}

// ---- degree count (incl. self loops) ---------------------------------------
__global__ void k_deg(const int* __restrict__ ei, float* __restrict__ deg) {
    int e = blockIdx.x * blockDim.x + threadIdx.x;
    if (e >= N_EDGES_TOT) return;
    int d, s; edge_pair(ei, e, d, s);
    atomicAdd(&deg[d], 1.0f);
}

// Stage kernel weights into LDS pair-transposed: lkT[(lev*NKP+kp)*N_FEAT+f] =
// ( kern[lev][2kp][f], kern[lev][2kp+1][f] )  -> B fragment = one ds_load_b64.
__device__ __forceinline__ void stage_weights(const float* __restrict__ kern,
                                              v2f* __restrict__ lkT,
                                              int levels, int tid, int nthreads) {
    int total = levels * NKP * N_FEAT;
    for (int p = tid; p < total; p += nthreads) {
        int f   = p & (N_FEAT - 1);
        int kp  = (p >> 5) & (NKP - 1);
        int lev = p >> 10;
        const float* kl = kern + lev * N_DIM * N_FEAT;
        v2f t;
        t[0] = kl[(2 * kp)     * N_FEAT + f];
        t[1] = kl[(2 * kp + 1) * N_FEAT + f];
        lkT[p] = t;
    }
}

// ---- Y[1] = x @ kernel[0] + bias[0]  (WMMA f32 16x16x4) --------------------
__global__ void __launch_bounds__(128) k_y1(const float* __restrict__ x,
                                            const float* __restrict__ kern,
                                            const float* __restrict__ bias,
                                            float* __restrict__ Ya) {
    __shared__ v2f   lkT[NKP * N_FEAT];    // kernel[0], pair-transposed: 8 KB
    __shared__ float lb[N_FEAT];
    int tid = threadIdx.x;
    stage_weights(kern, lkT, 1, tid, 128);
    if (tid < N_FEAT) lb[tid] = bias[tid];
    __syncthreads();

    int tile = blockIdx.x * 4 + (tid >> 5);
    if (tile >= N_NODES / 16) return;
    int lane = tid & 31;
    int h    = lane >> 4;     // half-wave: K-pair selector
    int mrow = lane & 15;     // A row / D column
    int nbase = tile * 16;

    // A fragments: x row (64 values over 16 k-steps x 2 vgprs)
    float a0[16], a1[16];
    const float2* xr = (const float2*)(x + (size_t)(nbase + mrow) * N_DIM);
#pragma unroll
    for (int kk = 0; kk < 16; ++kk) {
        float2 t = xr[kk * 2 + h];
        a0[kk] = t.x; a1[kk] = t.y;
    }

    for (int nt = 0; nt < 2; ++nt) {
        int f = nt * 16 + mrow;
        const v2f* lt = lkT + f;           // + kp*N_FEAT per k-step
        v8f acc = {};
#pragma unroll
        for (int kk = 0; kk < 16; ++kk) {
            int kp = kk * 2 + h;
            v2f A; A[0] = a0[kk]; A[1] = a1[kk];
            v2f B = lt[kp * N_FEAT];
            acc = __builtin_amdgcn_wmma_f32_16x16x4_f32(false, A, false, B,
                                                        (short)0, acc, false, false);
        }
        float bv = lb[f];
#pragma unroll
        for (int v = 0; v < 8; ++v) {
            int node = nbase + h * 8 + v;
            Ya[(size_t)NF + (size_t)node * N_FEAT + f] = acc[v] + bv;
        }
    }
}

// ---- per-step zero of Ynew levels 1..4 -------------------------------------
__global__ void k_zero(float* __restrict__ Ynew) {
    int i = blockIdx.x * blockDim.x + threadIdx.x;
    if (i < 4 * NF) Ynew[NF + i] = 0.0f;
}

// ---- main edge kernel: ell via WMMA, gather Y[src], scatter-add to Ynew ----
__global__ void __launch_bounds__(128) k_edge(const float* __restrict__ x,
                                              const int*   __restrict__ ei,
                                              const float* __restrict__ kern,
                                              const float* __restrict__ bias,
                                              const float* __restrict__ Ycur,
                                              float*       __restrict__ Ynew) {
    __shared__ v2f   lkT[N_LEV * NKP * N_FEAT];  // 4096 v2f = 32 KB
    __shared__ float lb[N_LEV * N_FEAT];         // 128 floats
    int tid = threadIdx.x;
    stage_weights(kern, lkT, N_LEV, tid, 128);
    if (tid < N_LEV * N_FEAT) lb[tid] = bias[tid];
    __syncthreads();

    int tile = blockIdx.x * 4 + (tid >> 5);
    if (tile >= N_EDGES_TOT / 16) return;
    int lane = tid & 31;
    int h    = lane >> 4;
    int mrow = lane & 15;
    int ebase = tile * 16;

    // A fragment: z = x[dst]-x[src] for edge row `mrow`
    int dA, sA; edge_pair(ei, ebase + mrow, dA, sA);
    const float2* xd = (const float2*)(x + (size_t)dA * N_DIM);
    const float2* xs = (const float2*)(x + (size_t)sA * N_DIM);
    float a0[16], a1[16];
#pragma unroll
    for (int kk = 0; kk < 16; ++kk) {
        float2 td = xd[kk * 2 + h];
        float2 ts = xs[kk * 2 + h];
        a0[kk] = td.x - ts.x;
        a1[kk] = td.y - ts.y;
    }

    // edge indices for the 8 D-rows this lane owns
    int dsts[8], srcs[8];
#pragma unroll
    for (int v = 0; v < 8; ++v)
        edge_pair(ei, ebase + h * 8 + v, dsts[v], srcs[v]);

    for (int lev = 0; lev < N_LEV; ++lev) {
        const float* Ylo  = Ycur + (size_t)lev * NF;
        const float* Yhi  = Ylo + NF;
        float*       Yout = Ynew + (size_t)(lev + 1) * NF;
        for (int nt = 0; nt < 2; ++nt) {
            int f = nt * 16 + mrow;
            const v2f* lt = lkT + lev * NKP * N_FEAT + f;
            v8f acc = {};
#pragma unroll
            for (int kk = 0; kk < 16; ++kk) {
                int kp = kk * 2 + h;
                v2f A; A[0] = a0[kk]; A[1] = a1[kk];
                v2f B = lt[kp * N_FEAT];
                acc = __builtin_amdgcn_wmma_f32_16x16x4_f32(false, A, false, B,
                                                            (short)0, acc, false, false);
            }
            float bv = lb[lev * N_FEAT + f];
#pragma unroll
            for (int v = 0; v < 8; ++v) {
                float ell = acc[v] + bv;          // D row = h*8+v, col = f
                int s = srcs[v], d = dsts[v];
                float ylo = Ylo[(size_t)s * N_FEAT + f];
                float yhi = Yhi[(size_t)s * N_FEAT + f];
                atomicAdd(&Yout[(size_t)d * N_FEAT + f], yhi + ylo * ell);
            }
        }
    }
}

// ---- mean normalization of levels 1..4 -------------------------------------
__global__ void k_norm(float* __restrict__ Ynew, const float* __restrict__ deg) {
    int i = blockIdx.x * blockDim.x + threadIdx.x;
    if (i >= 4 * NF) return;
    int node = (i % NF) / N_FEAT;
    Ynew[NF + i] *= (1.0f / deg[node]);
}

// ---- final: out[n,m,f] = Y[m+1][n,f] * embed_coeffs[m,f] -------------------
__global__ void k_out(const float* __restrict__ Ycur, const float* __restrict__ ec,
                      float* __restrict__ out) {
    int i = blockIdx.x * blockDim.x + threadIdx.x;
    if (i >= N_NODES * N_LEV * N_FEAT) return;
    int f = i & 31;
    int m = (i >> 5) & 3;
    int n = i >> 7;
    out[i] = Ycur[(size_t)(m + 1) * NF + (size_t)n * N_FEAT + f] * ec[m * N_FEAT + f];
}

extern "C" void kernel_launch(void* const* d_in, const int* in_sizes, int n_in,
                              void* d_out, int out_size, void* d_ws, size_t ws_size,
                              hipStream_t stream) {
    const float* x    = (const float*)d_in[0];
    const int*   ei   = (const int*)d_in[1];
    const float* kern = (const float*)d_in[2];
    const float* bias = (const float*)d_in[3];
    const float* ec   = (const float*)d_in[4];
    float*       out  = (float*)d_out;

    float* Ya  = (float*)d_ws;                 // 5*NF floats
    float* Yb  = Ya + (size_t)5 * NF;          // 5*NF floats
    float* deg = Yb + (size_t)5 * NF;          // N floats

    k_init<<<(5 * NF + 255) / 256, 256, 0, stream>>>(Ya, Yb, deg);
    k_deg<<<(N_EDGES_TOT + 255) / 256, 256, 0, stream>>>(ei, deg);
    k_y1<<<((N_NODES / 16) + 3) / 4, 128, 0, stream>>>(x, kern, bias, Ya);

    float* cur = Ya;
    float* nxt = Yb;
    for (int s = 0; s < N_STEPS; ++s) {
        k_zero<<<(4 * NF + 255) / 256, 256, 0, stream>>>(nxt);
        k_edge<<<((N_EDGES_TOT / 16) + 3) / 4, 128, 0, stream>>>(x, ei, kern, bias, cur, nxt);
        k_norm<<<(4 * NF + 255) / 256, 256, 0, stream>>>(nxt, deg);
        float* t = cur; cur = nxt; nxt = t;
    }
    k_out<<<(N_NODES * N_LEV * N_FEAT + 255) / 256, 256, 0, stream>>>(cur, ec, out);
}